// SimpleModel01_83305185673788
// MI455X (gfx1250) — compile-verified
//
#include <hip/hip_runtime.h>

typedef __attribute__((ext_vector_type(2))) float v2f;
typedef __attribute__((ext_vector_type(8))) float v8f;

#define NCELL 225
#define NPAD  240          // 15 WMMA tiles of 16 rows
#define WPB   8            // boards (waves) per 256-thread block
// per-wave LDS slice layout (bytes):
//   [0,768)      Wm   : f32 [12][16]  weight matrix (col0=policy,1=val+,2=val-)
//   [768,1728)   Lb   : f32 [240]
//   [1728,2688)  Pb   : f32 [240]
//   [2688,3648)  Mb   : f32 [240]
//   [3648,6528)  cnt8 : u8  [240][12] per-cell class counts (max 20)
//   [6528,6756)  sb8  : i8  [225]     board
//   [6756,7656)  lt   : u8  [900]     line class per (dir, center)
#define SLICE 7680

__device__ __forceinline__ float waveSum(float v) {
  #pragma unroll
  for (int m = 16; m > 0; m >>= 1) v += __shfl_xor(v, m, 32);
  return v;
}
__device__ __forceinline__ float waveMax(float v) {
  #pragma unroll
  for (int m = 16; m > 0; m >>= 1) v = fmaxf(v, __shfl_xor(v, m, 32));
  return v;
}

__global__ __launch_bounds__(32 * WPB) void gomoku_eval_kernel(
    const int*   __restrict__ state,     // (B,225) in {-1,0,1}
    const float* __restrict__ wpol_g,    // 11
    const float* __restrict__ wplus_g,   // 6
    const float* __restrict__ wmin_g,    // 6
    float*       __restrict__ out_logit, // (B,225)
    float*       __restrict__ out_value, // (B,)
    int nBoards)
{
  __shared__ __align__(16) unsigned char smem[WPB * SLICE];
  const int tid   = threadIdx.x;
  const int wave  = tid >> 5;
  const int lane  = tid & 31;
  const int hi    = lane >> 4;
  const int l     = lane & 15;
  const int board = blockIdx.x * WPB + wave;
  if (board >= nBoards) return;          // uniform per wave; no barriers exist below

  unsigned char* basep = smem + wave * SLICE;   // private slice: same-wave LDS is in-order
  float*         Wm   = (float*)(basep);
  float*         Lb   = (float*)(basep + 768);
  float*         Pb   = (float*)(basep + 1728);
  float*         Mb   = (float*)(basep + 2688);
  unsigned char* cnt8 = basep + 3648;
  signed char*   sb8  = (signed char*)(basep + 6528);
  unsigned char* lt   = basep + 6756;

  // ---- per-wave weight matrix (no cross-wave sharing -> no sync) ----
  for (int idx = lane; idx < 192; idx += 32) {
    int k = idx >> 4, n = idx & 15;
    float w = 0.f;
    if (n == 0)      w = (k >= 1)           ? wpol_g[k - 1]  : 0.f;  // classes 1..11
    else if (n == 1) w = (k >= 6)           ? wplus_g[k - 6] : 0.f;  // classes 6..11
    else if (n == 2) w = (k >= 1 && k <= 6) ? wmin_g[k - 1]  : 0.f;  // classes 1..6
    Wm[idx] = w;
  }

  // ---- board into LDS + zero the count rows ----
  #pragma unroll
  for (int t = 0; t < 8; ++t) {
    int i = lane + 32 * t;
    if (i < NCELL) sb8[i] = (signed char)state[(size_t)board * NCELL + i];
    if (i < NPAD) {
      unsigned int* z = (unsigned int*)(cnt8 + i * 12);
      z[0] = 0u; z[1] = 0u; z[2] = 0u;
    }
  }

  // ---- classify every 5-line: 4 dirs x 225 centers ----
  for (int e = lane; e < 4 * NCELL; e += 32) {
    int dir = e / NCELL; int p = e - dir * NCELL;
    int r = p / 15, c = p % 15;
    int dr, dc; bool valid;
    if (dir == 0)      { dr = 0; dc = 1;  valid = (c >= 2 && c <= 12); }
    else if (dir == 1) { dr = 1; dc = 0;  valid = (r >= 2 && r <= 12); }
    else if (dir == 2) { dr = 1; dc = 1;  valid = (r >= 2 && r <= 12 && c >= 2 && c <= 12); }
    else               { dr = 1; dc = -1; valid = (r >= 2 && r <= 12 && c >= 2 && c <= 12); }
    int ty = 0;
    if (valid) {
      int cp = 0, cm = 0;
      #pragma unroll
      for (int j = -2; j <= 2; ++j) {
        int v = sb8[(r + j * dr) * 15 + (c + j * dc)];
        cp += (v == 1);
        cm += (v == -1);
      }
      ty = (cp > 0 && cm > 0) ? 0 : (6 + cp - cm);   // dead/invalid -> class 0 (zero weight)
    }
    lt[e] = (unsigned char)ty;
  }

  // ---- per-cell class histogram (u8 in LDS; same-wave RMW is in-order) ----
  #pragma unroll
  for (int t = 0; t < 8; ++t) {
    int i = lane + 32 * t;
    if (i < NCELL) {
      int r = i / 15, c = i % 15;
      unsigned char* row = cnt8 + i * 12;
      #pragma unroll
      for (int j = -2; j <= 2; ++j) {                // horizontal center (r, c+j)
        int cc = c + j;
        if (cc >= 2 && cc <= 12) row[lt[0 * NCELL + r * 15 + cc]] += 1;
      }
      #pragma unroll
      for (int j = -2; j <= 2; ++j) {                // vertical center (r+j, c)
        int rr = r + j;
        if (rr >= 2 && rr <= 12) row[lt[1 * NCELL + rr * 15 + c]] += 1;
      }
      #pragma unroll
      for (int j = -2; j <= 2; ++j) {                // diag center (r+j, c+j)
        int rr = r + j, cc = c + j;
        if (rr >= 2 && rr <= 12 && cc >= 2 && cc <= 12)
          row[lt[2 * NCELL + rr * 15 + cc]] += 1;
      }
      #pragma unroll
      for (int j = -2; j <= 2; ++j) {                // anti-diag center (r+j, c-j)
        int rr = r + j, cc = c - j;
        if (rr >= 2 && rr <= 12 && cc >= 2 && cc <= 12)
          row[lt[3 * NCELL + rr * 15 + cc]] += 1;
      }
    }
  }

  // ---- (240x12)@(12x16) via 3x V_WMMA_F32_16X16X4_F32 per 16-row tile (f32 exact) ----
  // B 4x16 per chunk: VGPR0 = row K=2*hi, VGPR1 = row K=2*hi+1, col N = l
  v2f bv0 = { Wm[(0 + 2 * hi + 0) * 16 + l], Wm[(0 + 2 * hi + 1) * 16 + l] };
  v2f bv1 = { Wm[(4 + 2 * hi + 0) * 16 + l], Wm[(4 + 2 * hi + 1) * 16 + l] };
  v2f bv2 = { Wm[(8 + 2 * hi + 0) * 16 + l], Wm[(8 + 2 * hi + 1) * 16 + l] };
  for (int tile = 0; tile < NPAD / 16; ++tile) {
    const int mbase = tile * 16;
    const unsigned char* row = cnt8 + (mbase + l) * 12;  // A row M = l, K pair by lane half
    v2f a0 = { (float)row[0 + 2 * hi], (float)row[1 + 2 * hi] };
    v2f a1 = { (float)row[4 + 2 * hi], (float)row[5 + 2 * hi] };
    v2f a2 = { (float)row[8 + 2 * hi], (float)row[9 + 2 * hi] };
    v8f d = {};
    d = __builtin_amdgcn_wmma_f32_16x16x4_f32(false, a0, false, bv0, (short)0, d, false, false);
    d = __builtin_amdgcn_wmma_f32_16x16x4_f32(false, a1, false, bv1, (short)0, d, false, false);
    d = __builtin_amdgcn_wmma_f32_16x16x4_f32(false, a2, false, bv2, (short)0, d, false, false);
    if (l < 3) {                                        // D: row M=j+8*hi, col N=l
      float* dst = (l == 0) ? Lb : (l == 1) ? Pb : Mb;
      #pragma unroll
      for (int j = 0; j < 8; ++j) dst[mbase + j + 8 * hi] = d[j];
    }
  }

  // ---- epilogue: masked softmax, odds transform, tanh value (wave-local shuffles) ----
  float zeroA[8], logitA[8];
  float mx = -3.0e38f;
  #pragma unroll
  for (int t = 0; t < 8; ++t) {
    int i = lane + 32 * t;
    zeroA[t] = 0.f; logitA[t] = 0.f;
    if (i < NCELL) {
      float z  = (sb8[i] == 0) ? 1.f : 0.f;
      float lg = Lb[i] + 10000.f * z - 10000.f;
      zeroA[t] = z; logitA[t] = lg;
      out_logit[(size_t)board * NCELL + i] = lg;
      if (z > 0.5f) mx = fmaxf(mx, lg);
    }
  }
  mx = waveMax(mx);

  float exA[8]; float se = 0.f;
  #pragma unroll
  for (int t = 0; t < 8; ++t) {
    int i = lane + 32 * t;
    float e = (i < NCELL && zeroA[t] > 0.5f && mx > -1.0e37f) ? expf(logitA[t] - mx) : 0.f;
    exA[t] = e; se += e;
  }
  float S = fmaxf(waveSum(se), 1e-12f);

  float piA[8], odA[8]; float so = 0.f;
  #pragma unroll
  for (int t = 0; t < 8; ++t) {
    float p  = exA[t] / S;                 // softmax over empty cells, renormalized
    float pi = fminf(p, 0.99f);
    float od = pi / (1.f - pi);
    piA[t] = pi; odA[t] = od; so += od;
  }
  float SO = waveSum(so);

  float pvs = 0.f;
  #pragma unroll
  for (int t = 0; t < 8; ++t) {
    int i = lane + 32 * t;
    if (i < NCELL) {
      float pj = piA[t] * (SO - odA[t]);
      pvs += Pb[i] * piA[t] + Mb[i] * pj;
    }
  }
  float TV = waveSum(pvs);
  if (lane == 0) out_value[board] = tanhf(TV);
}

extern "C" void kernel_launch(void* const* d_in, const int* in_sizes, int n_in,
                              void* d_out, int out_size, void* d_ws, size_t ws_size,
                              hipStream_t stream) {
  const int*   state = (const int*)d_in[0];
  const float* wpol  = (const float*)d_in[1];   // 11 policy weights
  const float* wplus = (const float*)d_in[2];   // 6 value+ weights
  const float* wmin  = (const float*)d_in[3];   // 6 value- weights
  const int B = in_sizes[0] / NCELL;            // 4096 boards
  float* out = (float*)d_out;
  const int blocks = (B + WPB - 1) / WPB;
  gomoku_eval_kernel<<<blocks, 32 * WPB, 0, stream>>>(state, wpol, wplus, wmin,
                                                      out, out + (size_t)B * NCELL, B);
}